// HNM_70772471103666
// MI455X (gfx1250) — compile-verified
//
#include <hip/hip_runtime.h>

#define WD 14
#define NBLK 2048
#define NTHR 256

// workspace layout (in floats)
#define P_OFF     0      // head params: bR[14],cR,gR,sR0..2,gamR | bW..., erase[14]@40, add[14]@54
#define S_OFF     128    // scalars: maxR,maxW,sumER,sumEW,sumWR,sumWW
#define RH_OFF    160    // read_head accumulator (14)
#define PART_OFF  256    // 6*NBLK block partials
#define SCORE_OFF 16384  // scores/exp: 2*N floats

typedef __attribute__((ext_vector_type(2))) float v2f;
typedef __attribute__((ext_vector_type(4))) float v4f;
typedef __attribute__((ext_vector_type(8))) float v8f;

__device__ __forceinline__ float sigmoidf_(float x) { return 1.0f / (1.0f + expf(-x)); }
__device__ __forceinline__ float softplusf_(float x) { return x > 20.0f ? x : log1pf(expf(x)); }

// out[0..O) = act( in[0..K) @ W[KxO] + b[O] ), one full wave (32 lanes), f32 WMMA 16x16x4.
// act: 0 = tanh, 1 = identity. Only row M=0 of A is populated (batch of 1).
__device__ __forceinline__ void mlp_layer_wmma(const float* __restrict__ in,
                                               const float* __restrict__ W,
                                               const float* __restrict__ b,
                                               float* __restrict__ out,
                                               int K, int O, int act, int lane)
{
  const int m  = lane & 15;
  const int kh = lane >> 4;
  for (int t0 = 0; t0 < O; t0 += 16) {
    v8f c = {};
    const int col = t0 + m;
    for (int k0 = 0; k0 < K; k0 += 4) {
      const int ka = k0 + 2 * kh;   // A: lanes0-15 VGPR0=K0,VGPR1=K1; lanes16-31 K2,K3
      v2f a, bf;
      a.x  = (m == 0 && ka     < K) ? in[ka]     : 0.0f;
      a.y  = (m == 0 && ka + 1 < K) ? in[ka + 1] : 0.0f;
      bf.x = (ka     < K && col < O) ? W[(size_t)ka * O + col]       : 0.0f;
      bf.y = (ka + 1 < K && col < O) ? W[(size_t)(ka + 1) * O + col] : 0.0f;
      c = __builtin_amdgcn_wmma_f32_16x16x4_f32(false, a, false, bf, (short)0, c, false, false);
    }
    if (lane < 16 && col < O) {       // D row M=0 lives in VGPR0, lanes 0-15
      float v = c[0] + b[col];
      out[col] = act ? v : tanhf(v);
    }
  }
}

// ---- Kernel 0: MLP (WMMA) + head params -> ws[P_OFF..] --------------------
__global__ void k0_params(const float* __restrict__ X,
                          const float* __restrict__ W1, const float* __restrict__ b1,
                          const float* __restrict__ W2, const float* __restrict__ b2,
                          const float* __restrict__ W3, const float* __restrict__ b3,
                          const float* __restrict__ Wxi, const float* __restrict__ bxi,
                          float* __restrict__ ws)
{
  __shared__ float h1[32], h2[64], h3[128], xi[68];
  const int lane = threadIdx.x;
  mlp_layer_wmma(X,  W1,  b1,  h1, 14,  32, 0, lane); __syncthreads();
  mlp_layer_wmma(h1, W2,  b2,  h2, 32,  64, 0, lane); __syncthreads();
  mlp_layer_wmma(h2, W3,  b3,  h3, 64, 128, 0, lane); __syncthreads();
  mlp_layer_wmma(h3, Wxi, bxi, xi, 128, 68, 1, lane); __syncthreads();

  if (lane < 2) {  // lane 0: read head (rp=xi[0:20]); lane 1: write head (wp=xi[20:40])
    const float* p = xi + lane * 20;
    float* q = ws + P_OFF + lane * 20;
    float nrm = 0.0f;
    for (int j = 0; j < WD; ++j) {
      float kv = tanhf(p[j]) + 1e-16f;
      q[j] = kv; nrm += kv * kv;
    }
    nrm = fmaxf(sqrtf(nrm), 1e-8f);                 // nb = max(||k+eps||, EPS_COS)
    float g = sigmoidf_(p[WD]);
    float mx = fmaxf(p[15], fmaxf(p[16], p[17]));
    float e0 = expf(p[15] - mx), e1 = expf(p[16] - mx), e2 = expf(p[17] - mx);
    float es = e0 + e1 + e2;
    q[14] = softplusf_(p[19]) / nrm;                // c = beta / nb
    q[15] = g;
    q[16] = e0 / es; q[17] = e1 / es; q[18] = e2 / es;
    q[19] = 1.0f + softplusf_(p[18]);               // gamma
  }
  if (lane < WD) {
    ws[P_OFF + 40 + lane] = sigmoidf_(xi[40 + lane]);  // erase
    ws[P_OFF + 54 + lane] = tanhf(xi[54 + lane]);      // add
  }
}

// ---- Kernel 1: cosine scores for both heads + per-block max ----------------
__global__ void __launch_bounds__(NTHR) scores_kernel(const float* __restrict__ Mem,
                                                      float* __restrict__ ws, int N)
{
  __shared__ float pb[40];
  __shared__ float redR[NTHR], redW[NTHR];
  const int tid = threadIdx.x;
  if (tid < 40) pb[tid] = ws[P_OFF + tid];
  __syncthreads();
  float* scR = ws + SCORE_OFF;
  float* scW = scR + N;
  float mR = -1e30f, mW = -1e30f;
  const int stride = gridDim.x * blockDim.x;
  const int g0 = blockIdx.x * blockDim.x + tid;

  if ((N & 1) == 0) {
    // two rows / iteration: 112B chunk is 16B-aligned -> b128 loads
    const int NP = N >> 1;
    for (int p = g0; p < NP; p += stride) {
      float f[28];
      const v4f* mp = (const v4f*)(Mem + (size_t)p * 28);
#pragma unroll
      for (int t = 0; t < 7; ++t) {
        v4f v = mp[t];
        f[4*t] = v.x; f[4*t+1] = v.y; f[4*t+2] = v.z; f[4*t+3] = v.w;
      }
      v2f oR, oW;
#pragma unroll
      for (int r = 0; r < 2; ++r) {
        float dR = 0.0f, dW = 0.0f, nn = 0.0f;
#pragma unroll
        for (int j = 0; j < WD; ++j) {
          float a = f[r * WD + j] + 1e-16f;
          dR += a * pb[j]; dW += a * pb[20 + j]; nn += a * a;
        }
        float inv_na = 1.0f / fmaxf(sqrtf(nn), 1e-8f);
        float sR = dR * pb[14] * inv_na;
        float sW = dW * pb[34] * inv_na;
        if (r == 0) { oR.x = sR; oW.x = sW; } else { oR.y = sR; oW.y = sW; }
        mR = fmaxf(mR, sR); mW = fmaxf(mW, sW);
      }
      *(v2f*)(scR + (size_t)p * 2) = oR;
      *(v2f*)(scW + (size_t)p * 2) = oW;
    }
  } else {
    for (int i = g0; i < N; i += stride) {
      const float* row = Mem + (size_t)i * WD;
      float dR = 0.0f, dW = 0.0f, nn = 0.0f;
#pragma unroll
      for (int j = 0; j < WD; ++j) {
        float a = row[j] + 1e-16f;
        dR += a * pb[j]; dW += a * pb[20 + j]; nn += a * a;
      }
      float inv_na = 1.0f / fmaxf(sqrtf(nn), 1e-8f);
      float sR = dR * pb[14] * inv_na;
      float sW = dW * pb[34] * inv_na;
      scR[i] = sR; scW[i] = sW;
      mR = fmaxf(mR, sR); mW = fmaxf(mW, sW);
    }
  }
  redR[tid] = mR; redW[tid] = mW; __syncthreads();
  for (int s = NTHR / 2; s > 0; s >>= 1) {
    if (tid < s) { redR[tid] = fmaxf(redR[tid], redR[tid + s]); redW[tid] = fmaxf(redW[tid], redW[tid + s]); }
    __syncthreads();
  }
  if (tid == 0) { ws[PART_OFF + blockIdx.x] = redR[0]; ws[PART_OFF + NBLK + blockIdx.x] = redW[0]; }
}

// ---- Partial reduction: stage 0 = max pair, stage 1/2 = sum pair -----------
__global__ void reduce_kernel(float* __restrict__ ws, int stage)
{
  __shared__ float sA[NTHR], sB[NTHR];
  const int tid = threadIdx.x;
  const float* pa = ws + PART_OFF + (size_t)(2 * stage) * NBLK;
  const float* pc = pa + NBLK;
  float a = (stage == 0) ? -1e30f : 0.0f, b = a;
  for (int i = tid; i < NBLK; i += NTHR) {
    if (stage == 0) { a = fmaxf(a, pa[i]); b = fmaxf(b, pc[i]); }
    else            { a += pa[i];          b += pc[i]; }
  }
  sA[tid] = a; sB[tid] = b; __syncthreads();
  for (int s = NTHR / 2; s > 0; s >>= 1) {
    if (tid < s) {
      if (stage == 0) { sA[tid] = fmaxf(sA[tid], sA[tid + s]); sB[tid] = fmaxf(sB[tid], sB[tid + s]); }
      else            { sA[tid] += sA[tid + s];                sB[tid] += sB[tid + s]; }
    }
    __syncthreads();
  }
  if (tid == 0) { ws[S_OFF + 2 * stage] = sA[0]; ws[S_OFF + 2 * stage + 1] = sB[0]; }
  if (stage == 2 && tid < WD) ws[RH_OFF + tid] = 0.0f;   // zero read-head accumulator
}

// ---- Kernel 2: e = exp(score - max), in place, + per-block sums ------------
__global__ void __launch_bounds__(NTHR) expsum_kernel(float* __restrict__ ws, int N)
{
  __shared__ float redR[NTHR], redW[NTHR];
  const int tid = threadIdx.x;
  const float maxR = ws[S_OFF + 0], maxW = ws[S_OFF + 1];
  float* scR = ws + SCORE_OFF;
  float* scW = scR + N;
  float aR = 0.0f, aW = 0.0f;
  const int stride = gridDim.x * blockDim.x;
  const int g0 = blockIdx.x * blockDim.x + tid;

  if ((N & 3) == 0) {
    const int NQ = N >> 2;
    v4f* r4 = (v4f*)scR;
    v4f* w4 = (v4f*)scW;
    for (int q = g0; q < NQ; q += stride) {
      v4f a = r4[q];
      a.x = expf(a.x - maxR); a.y = expf(a.y - maxR);
      a.z = expf(a.z - maxR); a.w = expf(a.w - maxR);
      r4[q] = a; aR += (a.x + a.y) + (a.z + a.w);
      v4f b = w4[q];
      b.x = expf(b.x - maxW); b.y = expf(b.y - maxW);
      b.z = expf(b.z - maxW); b.w = expf(b.w - maxW);
      w4[q] = b; aW += (b.x + b.y) + (b.z + b.w);
    }
  } else {
    for (int i = g0; i < N; i += stride) {
      float eR = expf(scR[i] - maxR);
      float eW = expf(scW[i] - maxW);
      scR[i] = eR; scW[i] = eW;
      aR += eR; aW += eW;
    }
  }
  redR[tid] = aR; redW[tid] = aW; __syncthreads();
  for (int s = NTHR / 2; s > 0; s >>= 1) {
    if (tid < s) { redR[tid] += redR[tid + s]; redW[tid] += redW[tid + s]; }
    __syncthreads();
  }
  if (tid == 0) { ws[PART_OFF + 2 * NBLK + blockIdx.x] = redR[0]; ws[PART_OFF + 3 * NBLK + blockIdx.x] = redW[0]; }
}

// ---- Kernel 3: gate + circular shift + sharpen, unnormalized, + sums -------
__global__ void __launch_bounds__(NTHR) w_kernel(const float* __restrict__ rprev,
                                                 const float* __restrict__ wprev,
                                                 float* __restrict__ ws,
                                                 float* __restrict__ out_rw,
                                                 float* __restrict__ out_ww, int N)
{
  __shared__ float redR[NTHR], redW[NTHR];
  const int tid = threadIdx.x;
  const float gR = ws[P_OFF + 15], s0R = ws[P_OFF + 16], s1R = ws[P_OFF + 17], s2R = ws[P_OFF + 18], gaR = ws[P_OFF + 19];
  const float gW = ws[P_OFF + 35], s0W = ws[P_OFF + 36], s1W = ws[P_OFF + 37], s2W = ws[P_OFF + 38], gaW = ws[P_OFF + 39];
  const float invER = 1.0f / ws[S_OFF + 2];
  const float invEW = 1.0f / ws[S_OFF + 3];
  const float* eR = ws + SCORE_OFF;
  const float* eW = eR + N;
  float sumR = 0.0f, sumW = 0.0f;
  const int stride = gridDim.x * blockDim.x;
  const int g0 = blockIdx.x * blockDim.x + tid;

  if ((N & 3) == 0) {
    const int NQ = N >> 2;
    for (int q = g0; q < NQ; q += stride) {
      const int i = q << 2;
      const int im = (i == 0) ? N - 1 : i - 1;
      const int ip = (i + 4 == N) ? 0 : i + 4;
      // ---- read head ----
      {
        v4f e4 = *(const v4f*)(eR + i);
        v4f p4 = *(const v4f*)(rprev + i);
        float wg[6];
        wg[0] = gR * eR[im] * invER + (1.0f - gR) * rprev[im];
        wg[1] = gR * e4.x  * invER + (1.0f - gR) * p4.x;
        wg[2] = gR * e4.y  * invER + (1.0f - gR) * p4.y;
        wg[3] = gR * e4.z  * invER + (1.0f - gR) * p4.z;
        wg[4] = gR * e4.w  * invER + (1.0f - gR) * p4.w;
        wg[5] = gR * eR[ip] * invER + (1.0f - gR) * rprev[ip];
        v4f o;
        o.x = powf(s0R * wg[0] + s1R * wg[1] + s2R * wg[2], gaR);
        o.y = powf(s0R * wg[1] + s1R * wg[2] + s2R * wg[3], gaR);
        o.z = powf(s0R * wg[2] + s1R * wg[3] + s2R * wg[4], gaR);
        o.w = powf(s0R * wg[3] + s1R * wg[4] + s2R * wg[5], gaR);
        *(v4f*)(out_rw + i) = o;
        sumR += (o.x + o.y) + (o.z + o.w);
      }
      // ---- write head ----
      {
        v4f e4 = *(const v4f*)(eW + i);
        v4f p4 = *(const v4f*)(wprev + i);
        float wg[6];
        wg[0] = gW * eW[im] * invEW + (1.0f - gW) * wprev[im];
        wg[1] = gW * e4.x  * invEW + (1.0f - gW) * p4.x;
        wg[2] = gW * e4.y  * invEW + (1.0f - gW) * p4.y;
        wg[3] = gW * e4.z  * invEW + (1.0f - gW) * p4.z;
        wg[4] = gW * e4.w  * invEW + (1.0f - gW) * p4.w;
        wg[5] = gW * eW[ip] * invEW + (1.0f - gW) * wprev[ip];
        v4f o;
        o.x = powf(s0W * wg[0] + s1W * wg[1] + s2W * wg[2], gaW);
        o.y = powf(s0W * wg[1] + s1W * wg[2] + s2W * wg[3], gaW);
        o.z = powf(s0W * wg[2] + s1W * wg[3] + s2W * wg[4], gaW);
        o.w = powf(s0W * wg[3] + s1W * wg[4] + s2W * wg[5], gaW);
        *(v4f*)(out_ww + i) = o;
        sumW += (o.x + o.y) + (o.z + o.w);
      }
    }
  } else {
    for (int i = g0; i < N; i += stride) {
      const int im = (i == 0) ? N - 1 : i - 1;
      const int ip = (i == N - 1) ? 0 : i + 1;
      float wgm = gR * eR[im] * invER + (1.0f - gR) * rprev[im];
      float wg0 = gR * eR[i]  * invER + (1.0f - gR) * rprev[i];
      float wgp = gR * eR[ip] * invER + (1.0f - gR) * rprev[ip];
      float w = powf(s0R * wgm + s1R * wg0 + s2R * wgp, gaR);
      out_rw[i] = w; sumR += w;
      wgm = gW * eW[im] * invEW + (1.0f - gW) * wprev[im];
      wg0 = gW * eW[i]  * invEW + (1.0f - gW) * wprev[i];
      wgp = gW * eW[ip] * invEW + (1.0f - gW) * wprev[ip];
      w = powf(s0W * wgm + s1W * wg0 + s2W * wgp, gaW);
      out_ww[i] = w; sumW += w;
    }
  }
  redR[tid] = sumR; redW[tid] = sumW; __syncthreads();
  for (int s = NTHR / 2; s > 0; s >>= 1) {
    if (tid < s) { redR[tid] += redR[tid + s]; redW[tid] += redW[tid + s]; }
    __syncthreads();
  }
  if (tid == 0) { ws[PART_OFF + 4 * NBLK + blockIdx.x] = redR[0]; ws[PART_OFF + 5 * NBLK + blockIdx.x] = redW[0]; }
}

// ---- Kernel 4: normalize weights, memory update, read_head partials --------
__global__ void __launch_bounds__(NTHR) finalize_kernel(const float* __restrict__ Mem,
                                                        float* __restrict__ ws,
                                                        float* __restrict__ out_rw,
                                                        float* __restrict__ out_ww,
                                                        float* __restrict__ out_mem, int N)
{
  __shared__ float er[WD], ad[WD];
  __shared__ float red[NTHR * WD];     // 14 KB
  const int tid = threadIdx.x;
  if (tid < WD) { er[tid] = ws[P_OFF + 40 + tid]; ad[tid] = ws[P_OFF + 54 + tid]; }
  __syncthreads();
  const float invWR = 1.0f / (ws[S_OFF + 4] + 1e-16f);
  const float invWW = 1.0f / (ws[S_OFF + 5] + 1e-16f);
  float acc[WD];
#pragma unroll
  for (int j = 0; j < WD; ++j) acc[j] = 0.0f;
  const int stride = gridDim.x * blockDim.x;
  const int g0 = blockIdx.x * blockDim.x + tid;

  if ((N & 1) == 0) {
    const int NP = N >> 1;
    for (int p = g0; p < NP; p += stride) {
      v2f rw2 = *(const v2f*)(out_rw + (size_t)p * 2);
      v2f ww2 = *(const v2f*)(out_ww + (size_t)p * 2);
      rw2.x *= invWR; rw2.y *= invWR;
      ww2.x *= invWW; ww2.y *= invWW;
      *(v2f*)(out_rw + (size_t)p * 2) = rw2;
      *(v2f*)(out_ww + (size_t)p * 2) = ww2;
      const float rws[2] = { rw2.x, rw2.y };
      const float wvs[2] = { ww2.x, ww2.y };
      float f[28], gm[28];
      const v4f* mp = (const v4f*)(Mem + (size_t)p * 28);
#pragma unroll
      for (int t = 0; t < 7; ++t) {
        v4f v = mp[t];
        f[4*t] = v.x; f[4*t+1] = v.y; f[4*t+2] = v.z; f[4*t+3] = v.w;
      }
#pragma unroll
      for (int r = 0; r < 2; ++r)
#pragma unroll
        for (int j = 0; j < WD; ++j) {
          float mn = f[r * WD + j] * (1.0f - wvs[r] * er[j]) + wvs[r] * ad[j];
          gm[r * WD + j] = mn;
          acc[j] += rws[r] * mn;
        }
      v4f* op = (v4f*)(out_mem + (size_t)p * 28);
#pragma unroll
      for (int t = 0; t < 7; ++t) {
        v4f v;
        v.x = gm[4*t]; v.y = gm[4*t+1]; v.z = gm[4*t+2]; v.w = gm[4*t+3];
        __builtin_nontemporal_store(v, op + t);   // Memory_new never re-read: keep out of L2
      }
    }
  } else {
    for (int i = g0; i < N; i += stride) {
      float rw = out_rw[i] * invWR;
      float wv = out_ww[i] * invWW;
      out_rw[i] = rw; out_ww[i] = wv;
      const float* row = Mem + (size_t)i * WD;
      float* orow = out_mem + (size_t)i * WD;
#pragma unroll
      for (int j = 0; j < WD; ++j) {
        float mn = row[j] * (1.0f - wv * er[j]) + wv * ad[j];
        orow[j] = mn;
        acc[j] += rw * mn;
      }
    }
  }
#pragma unroll
  for (int j = 0; j < WD; ++j) red[tid * WD + j] = acc[j];
  __syncthreads();
  for (int s = NTHR / 2; s > 0; s >>= 1) {
    if (tid < s)
      for (int j = 0; j < WD; ++j) red[tid * WD + j] += red[(tid + s) * WD + j];
    __syncthreads();
  }
  if (tid < WD) atomicAdd(&ws[RH_OFF + tid], red[tid]);
}

// ---- Kernel 5: P-network + mixture-of-nets output (8 floats) ---------------
__global__ void final_small(const float* __restrict__ X,
                            const float* __restrict__ Wz1, const float* __restrict__ bz1,
                            const float* __restrict__ Wz2, const float* __restrict__ bz2,
                            const float* __restrict__ netW, const float* __restrict__ netb,
                            const float* __restrict__ ws, float* __restrict__ out)
{
  __shared__ float alu[28], P1[16], P2[3], nets[24];
  const int tid = threadIdx.x;
  if (tid < 14) { alu[tid] = X[tid]; alu[14 + tid] = ws[RH_OFF + tid]; }
  __syncthreads();
  if (tid < 16) {
    float v = bz1[tid];
    for (int i = 0; i < 28; ++i) v += alu[i] * Wz1[i * 16 + tid];
    P1[tid] = tanhf(v);
  }
  __syncthreads();
  if (tid == 0) {
    float l0 = bz2[0], l1 = bz2[1], l2 = bz2[2];
    for (int i = 0; i < 16; ++i) { l0 += P1[i] * Wz2[i * 3]; l1 += P1[i] * Wz2[i * 3 + 1]; l2 += P1[i] * Wz2[i * 3 + 2]; }
    float m = fmaxf(l0, fmaxf(l1, l2));
    float e0 = expf(l0 - m), e1 = expf(l1 - m), e2 = expf(l2 - m);
    float es = e0 + e1 + e2;
    P2[0] = e0 / es; P2[1] = e1 / es; P2[2] = e2 / es;
  }
  __syncthreads();
  if (tid < 24) {
    int k = tid >> 3, o = tid & 7;
    float v = netb[k * 8 + o];
    for (int i = 0; i < 8; ++i) v += X[i] * netW[k * 64 + i * 8 + o];
    nets[tid] = sigmoidf_(v);
  }
  __syncthreads();
  if (tid < 8) out[tid] = P2[0] * nets[tid] + P2[1] * nets[8 + tid] + P2[2] * nets[16 + tid];
}

extern "C" void kernel_launch(void* const* d_in, const int* in_sizes, int n_in,
                              void* d_out, int out_size, void* d_ws, size_t ws_size,
                              hipStream_t stream) {
  const float* X     = (const float*)d_in[0];
  const float* rprev = (const float*)d_in[1];
  const float* wprev = (const float*)d_in[2];
  const float* Mem   = (const float*)d_in[3];
  const float* W1  = (const float*)d_in[4];  const float* b1  = (const float*)d_in[5];
  const float* W2  = (const float*)d_in[6];  const float* b2  = (const float*)d_in[7];
  const float* W3  = (const float*)d_in[8];  const float* b3  = (const float*)d_in[9];
  const float* Wxi = (const float*)d_in[10]; const float* bxi = (const float*)d_in[11];
  const float* Wz1 = (const float*)d_in[12]; const float* bz1 = (const float*)d_in[13];
  const float* Wz2 = (const float*)d_in[14]; const float* bz2 = (const float*)d_in[15];
  const float* netW = (const float*)d_in[16]; const float* netb = (const float*)d_in[17];

  const int N = in_sizes[1];              // 4,000,000
  float* out = (float*)d_out;
  float* ws  = (float*)d_ws;
  float* out_rw  = out + 8;
  float* out_ww  = out + 8 + (size_t)N;
  float* out_mem = out + 8 + 2 * (size_t)N;

  k0_params<<<1, 32, 0, stream>>>(X, W1, b1, W2, b2, W3, b3, Wxi, bxi, ws);
  scores_kernel<<<NBLK, NTHR, 0, stream>>>(Mem, ws, N);
  reduce_kernel<<<1, NTHR, 0, stream>>>(ws, 0);
  expsum_kernel<<<NBLK, NTHR, 0, stream>>>(ws, N);
  reduce_kernel<<<1, NTHR, 0, stream>>>(ws, 1);
  w_kernel<<<NBLK, NTHR, 0, stream>>>(rprev, wprev, ws, out_rw, out_ww, N);
  reduce_kernel<<<1, NTHR, 0, stream>>>(ws, 2);
  finalize_kernel<<<NBLK, NTHR, 0, stream>>>(Mem, ws, out_rw, out_ww, out_mem, N);
  final_small<<<1, 32, 0, stream>>>(X, Wz1, bz1, Wz2, bz2, netW, netb, ws, out);
}